// GAT_15195594293519
// MI455X (gfx1250) — compile-verified
//
#include <hip/hip_runtime.h>

#define DIMS_IN  128
#define DIMS_H   96
#define DIMS_OUT 32

typedef float v2f __attribute__((ext_vector_type(2)));
typedef float v8f __attribute__((ext_vector_type(8)));

// ---------------------------------------------------------------------------
// Zero-fill (float4 granularity; all our buffers are multiples of 16 bytes)
// ---------------------------------------------------------------------------
__global__ void zero_kernel(float4* __restrict__ p, int n4) {
    int i = blockIdx.x * blockDim.x + threadIdx.x;
    if (i < n4) {
        float4 z; z.x = 0.f; z.y = 0.f; z.z = 0.f; z.w = 0.f;
        p[i] = z;
    }
}

// ---------------------------------------------------------------------------
// Tiny weight transpose: Wt[n*K + k] = W[k*N + n]   (W is [K x N] row-major)
// W1: 128x96 (48KB), W2: 96x32 (12KB) -- negligible cost, done once per call.
// ---------------------------------------------------------------------------
__global__ void transpose_kernel(const float* __restrict__ W,
                                 float* __restrict__ Wt, int K, int N) {
    int i = blockIdx.x * blockDim.x + threadIdx.x;
    if (i < K * N) {
        int k = i / N;
        int n = i - k * N;
        Wt[n * K + k] = W[i];
    }
}

// ---------------------------------------------------------------------------
// Dense GEMM C[MxN] = op(A)[MxK] * B[KxN] with B given TRANSPOSED ([N x K]
// row-major). fp32 WMMA (V_WMMA_F32_16X16X4_F32), one wave per 16x32 output
// strip (two 16x16 tiles sharing the A fragment).
//
// Per-lane fragments (ISA 7.12.2, 32-bit A 16x4 / B 4x16 layouts):
//   A: lane l (0-15) holds row M=l, K={k,k+1}; lanes 16-31 hold K={k+2,k+3}
//      -> contiguous b64 load from A + row*K + half*2 + k   (immediate offsets)
//   B: lane l (0-15) holds col N=l, K={k,k+1}; lanes 16-31 hold K={k+2,k+3}
//      -> contiguous b64 load from Bt + col*K + half*2 + k  (immediate offsets)
//   C/D: VGPR r -> M=r (lanes 0-15) / M=r+8 (lanes 16-31), N = lane&15
//
// RELU_A fuses layer-1 activation into the A-operand load.
// Requires M%16==0, N%32==0, K%4==0 (holds: M=50000, N=96/32, K=128/96).
// ---------------------------------------------------------------------------
template <int K, bool RELU_A>
__global__ void gemm_wmma_f32_bt(const float* __restrict__ A,
                                 const float* __restrict__ Bt,
                                 float* __restrict__ C,
                                 int M, int N) {
    const int wavesPerBlock = blockDim.x >> 5;
    const int waveId = blockIdx.x * wavesPerBlock + (threadIdx.x >> 5);
    const int lane   = threadIdx.x & 31;
    const int nPair  = N >> 5;                       // 32-column strips
    const int tileM  = waveId / nPair;
    const int tileP  = waveId - tileM * nPair;
    if (tileM * 16 >= M) return;                     // wave-uniform exit

    const int half = lane >> 4;                      // K sub-pair selector
    const int l15  = lane & 15;

    // Loop-invariant base pointers; all k-offsets become immediates.
    const float* arow = A  + (tileM * 16 + l15) * K + half * 2;
    const float* brow = Bt + (tileP * 32 + l15) * K + half * 2;

    v8f acc0 = {};
    v8f acc1 = {};
    #pragma unroll 4
    for (int k = 0; k < K; k += 4) {
        v2f a = *(const v2f*)(arow + k);
        if (RELU_A) {
            a.x = fmaxf(a.x, 0.f);
            a.y = fmaxf(a.y, 0.f);
        }
        v2f b0 = *(const v2f*)(brow + k);            // tile tileP*2
        v2f b1 = *(const v2f*)(brow + 16 * K + k);   // tile tileP*2+1 (+8KB max)
        // (neg_a, A, neg_b, B, c_mod, C, reuse_a, reuse_b)
        acc0 = __builtin_amdgcn_wmma_f32_16x16x4_f32(
            false, a, false, b0, (short)0, acc0, false, false);
        acc1 = __builtin_amdgcn_wmma_f32_16x16x4_f32(
            false, a, false, b1, (short)0, acc1, false, false);
    }

    float* c0 = C + (tileM * 16 + half * 8) * N + tileP * 32 + l15;
    #pragma unroll
    for (int r = 0; r < 8; ++r) {
        c0[r * N]      = acc0[r];
        c0[r * N + 16] = acc1[r];
    }
}

// ---------------------------------------------------------------------------
// COO SpMM scatter: out[dst[e]] += val[e] * dense[src[e]]  (atomic f32 adds)
// One thread per (edge, 4-feature chunk); 16B gathers hit L2 (dense matrix is
// 19.2MB << 192MB L2), 4x global_atomic_add_f32 per thread.
// ---------------------------------------------------------------------------
__global__ void spmm_scatter(const int* __restrict__ esrc,
                             const int* __restrict__ edst,
                             const float* __restrict__ eval,
                             const float* __restrict__ dense,
                             float* __restrict__ out,
                             int total, int D4, int D) {
    int i = blockIdx.x * blockDim.x + threadIdx.x;
    if (i >= total) return;
    int e = i / D4;
    int c = (i - e * D4) * 4;
    int s = esrc[e];
    int d = edst[e];
    float v = eval[e];
    const float4 m = *(const float4*)(dense + (size_t)s * D + c);
    float* o = out + (size_t)d * D + c;
    atomicAdd(o + 0, v * m.x);
    atomicAdd(o + 1, v * m.y);
    atomicAdd(o + 2, v * m.z);
    atomicAdd(o + 3, v * m.w);
}

// ---------------------------------------------------------------------------
// Orchestration
// ---------------------------------------------------------------------------
extern "C" void kernel_launch(void* const* d_in, const int* in_sizes, int n_in,
                              void* d_out, int out_size, void* d_ws, size_t ws_size,
                              hipStream_t stream) {
    const float* x    = (const float*)d_in[0];
    const int*   esrc = (const int*)d_in[1];
    const int*   edst = (const int*)d_in[2];
    const float* eval = (const float*)d_in[3];
    const float* W1   = (const float*)d_in[4];
    const float* W2   = (const float*)d_in[5];
    float*       out  = (float*)d_out;

    const int E = in_sizes[1];
    const int n = in_sizes[0] / DIMS_IN;          // number of nodes (50000)

    // Workspace layout
    float* buf0 = (float*)d_ws;                   // n*DIMS_H  : XW1, later HW2
    float* buf1 = buf0 + (size_t)n * DIMS_H;      // n*DIMS_H  : SpMM1 accumulator
    float* W1t  = buf1 + (size_t)n * DIMS_H;      // 96*128 transposed W1
    float* W2t  = W1t + DIMS_H * DIMS_IN;         // 32*96  transposed W2

    // 0) transpose weights (independent of everything else)
    transpose_kernel<<<(DIMS_IN * DIMS_H + 255) / 256, 256, 0, stream>>>(
        W1, W1t, DIMS_IN, DIMS_H);
    transpose_kernel<<<(DIMS_H * DIMS_OUT + 255) / 256, 256, 0, stream>>>(
        W2, W2t, DIMS_H, DIMS_OUT);

    // 1) zero SpMM accumulators (buf1 and d_out)
    {
        int n4 = (n * DIMS_H) / 4;
        zero_kernel<<<(n4 + 255) / 256, 256, 0, stream>>>((float4*)buf1, n4);
        int o4 = (n * DIMS_OUT) / 4;
        zero_kernel<<<(o4 + 255) / 256, 256, 0, stream>>>((float4*)out, o4);
    }

    // 2) GEMM1: buf0 = x @ W1   [n x 128] @ [128 x 96]
    {
        int waves  = (n / 16) * (DIMS_H / 32);    // 3125 * 3
        int blocks = (waves + 7) / 8;             // 8 waves per 256-thread block
        gemm_wmma_f32_bt<DIMS_IN, false><<<blocks, 256, 0, stream>>>(
            x, W1t, buf0, n, DIMS_H);
    }

    // 3) SpMM1: buf1 += scatter(edge_val * buf0[src])
    {
        int total = E * (DIMS_H / 4);
        spmm_scatter<<<(total + 255) / 256, 256, 0, stream>>>(
            esrc, edst, eval, buf0, buf1, total, DIMS_H / 4, DIMS_H);
    }

    // 4) GEMM2: buf0 = relu(buf1) @ W2   [n x 96] @ [96 x 32]  (relu fused on A)
    {
        int waves  = (n / 16) * (DIMS_OUT / 32);  // 3125
        int blocks = (waves + 7) / 8;
        gemm_wmma_f32_bt<DIMS_H, true><<<blocks, 256, 0, stream>>>(
            buf1, W2t, buf0, n, DIMS_OUT);
    }

    // 5) SpMM2: d_out += scatter(edge_val * buf0[src])
    {
        int total = E * (DIMS_OUT / 4);
        spmm_scatter<<<(total + 255) / 256, 256, 0, stream>>>(
            esrc, edst, eval, buf0, out, total, DIMS_OUT / 4, DIMS_OUT);
    }
}